// GCN_20426864460528
// MI455X (gfx1250) — compile-verified
//
#include <hip/hip_runtime.h>
#include <hip/hip_bf16.h>

// ---------------------------------------------------------------------------
// GCN (2x GCNConv + gather-sum pooling + 3-layer MLP head), fp32 end-to-end.
// GEMMs use CDNA5 native fp32 WMMA: v_wmma_f32_16x16x4_f32, with each wave
// computing a 16x64 output strip (4 accumulators) to amortize A-fragment
// loads. Edge aggregation uses b128 gathers + hardware global_atomic_add_f32.
// ---------------------------------------------------------------------------

#define GN 50000
#define GE 800000
#define GF 256
#define GB 1024
#define GL 50

typedef __attribute__((ext_vector_type(2))) float v2f;
typedef __attribute__((ext_vector_type(8))) float v8f;

__device__ __forceinline__ void atom_add_f32(float* p, float v) {
    __hip_atomic_fetch_add(p, v, __ATOMIC_RELAXED, __HIP_MEMORY_SCOPE_AGENT);
}

// ---------------- degree / normalization ----------------
__global__ void k_deg_init(float* deg, int n) {
    int i = blockIdx.x * blockDim.x + threadIdx.x;
    if (i < n) deg[i] = 1.0f;  // self loop
}

__global__ void k_deg_accum(const int* __restrict__ dst, float* __restrict__ deg, int e) {
    int i = blockIdx.x * blockDim.x + threadIdx.x;
    if (i < e) atom_add_f32(&deg[dst[i]], 1.0f);
}

__global__ void k_deg_to_dinv(float* deg, int n) {
    int i = blockIdx.x * blockDim.x + threadIdx.x;
    if (i < n) deg[i] = rsqrtf(deg[i]);
}

// ---------------- fp32 WMMA GEMM: C[M,Nc] = A[M,K] @ W[K,Nc] (+bias)(relu) --
// One wave computes a 16x64 strip: 4 accumulators, stepping K by 4 with
// v_wmma_f32_16x16x4_f32, reusing the A fragment across the 4 column tiles.
// Block = 256 threads = 8 waves. strips_per_row = Nc/64; waves are split
// between (8/strips_per_row) row-blocks of 16 rows each.
// Layouts per CDNA5 ISA 7.12.2:
//   A 16x4 f32 : lanes 0-15 rows M=0..15 holding K=k0,k0+1 in v0,v1;
//                lanes 16-31 same rows holding K=k0+2,k0+3.
//   B 4x16 f32 : v0 lanes0-15 = row K=k0 (N across lanes), lanes16-31 = K=k0+2;
//                v1 = K=k0+1 / K=k0+3.
//   C/D        : VGPR r, lanes0-15 -> (M=r, N=lane); lanes16-31 -> (M=r+8).
__global__ void k_wmma_gemm_f32(const float* __restrict__ A,
                                const float* __restrict__ W,
                                const float* __restrict__ bias,
                                float* __restrict__ C,
                                int M, int K, int Nc, int relu) {
    const int lane = threadIdx.x & 31;
    const int wave = threadIdx.x >> 5;
    const int strips = Nc >> 6;          // 64-wide strips per row block (4 or 2)
    const int rows_per_blk = 8 / strips; // row-blocks of 16 rows per workgroup

    const int rowblk = wave / strips;
    const int strip  = wave % strips;
    const int m0 = (blockIdx.x * rows_per_blk + rowblk) * 16;
    const int n0 = strip * 64;
    if (m0 >= M) return;

    const int half = lane >> 4;  // 0: K pair (k0,k0+1), 1: (k0+2,k0+3)
    const int l16  = lane & 15;

    const float* __restrict__ arow = A + (size_t)(m0 + l16) * K;

    v8f acc0 = {}, acc1 = {}, acc2 = {}, acc3 = {};
    for (int k0 = 0; k0 < K; k0 += 4) {
        v2f a = *(const v2f*)(arow + k0 + half * 2);
        const float* __restrict__ w0 = W + (size_t)(k0 + half * 2) * Nc + n0 + l16;
        const float* __restrict__ w1 = w0 + Nc;
        v2f b0, b1, b2, b3;
        b0.x = w0[0];  b0.y = w1[0];
        b1.x = w0[16]; b1.y = w1[16];
        b2.x = w0[32]; b2.y = w1[32];
        b3.x = w0[48]; b3.y = w1[48];
        acc0 = __builtin_amdgcn_wmma_f32_16x16x4_f32(false, a, false, b0, (short)0, acc0, false, false);
        acc1 = __builtin_amdgcn_wmma_f32_16x16x4_f32(false, a, false, b1, (short)0, acc1, false, false);
        acc2 = __builtin_amdgcn_wmma_f32_16x16x4_f32(false, a, false, b2, (short)0, acc2, false, false);
        acc3 = __builtin_amdgcn_wmma_f32_16x16x4_f32(false, a, false, b3, (short)0, acc3, false, false);
    }

    float* __restrict__ crow = C + (size_t)(m0 + half * 8) * Nc + n0 + l16;
    v8f* accs[4] = {&acc0, &acc1, &acc2, &acc3};
#pragma unroll
    for (int t = 0; t < 4; ++t) {
        const float bv = bias ? bias[n0 + t * 16 + l16] : 0.0f;
        v8f acc = *accs[t];
#pragma unroll
        for (int r = 0; r < 8; ++r) {
            float v = acc[r] + bv;
            if (relu) v = fmaxf(v, 0.0f);
            crow[(size_t)r * Nc + t * 16] = v;
        }
    }
}

// ---------------- self-loop init: agg[i,:] = h[i,:] * dinv[i]^2 ------------
__global__ void k_self_init(const float* __restrict__ h,
                            const float* __restrict__ dinv,
                            float* __restrict__ agg, int n) {
    int i = blockIdx.x * blockDim.x + threadIdx.x;  // over N*F
    if (i < n) {
        int node = i >> 8;  // F == 256
        float d = dinv[node];
        agg[i] = h[i] * (d * d);
    }
}

// ---------------- edge scatter: agg[dst,:] += h[src,:] * dinv[s]*dinv[d] ---
// One wave per edge (8 edges per 256-thread block); each lane handles 8
// features via two float4 (b128) gathers + 8 f32 atomics.
__global__ void k_edge_agg(const float* __restrict__ h,
                           const int* __restrict__ src,
                           const int* __restrict__ dst,
                           const float* __restrict__ dinv,
                           float* __restrict__ agg, int e) {
    int edge = blockIdx.x * 8 + (threadIdx.x >> 5);
    if (edge >= e) return;
    int lane = threadIdx.x & 31;
    int s = src[edge];
    int d = dst[edge];
    float norm = dinv[s] * dinv[d];
    const float4* __restrict__ hp = (const float4*)(h + (size_t)s * GF);
    float* __restrict__ ap = agg + (size_t)d * GF;
#pragma unroll
    for (int j = 0; j < 2; ++j) {
        int vi = lane + j * 32;          // float4 index within the row
        float4 v = hp[vi];
        int base = vi * 4;
        atom_add_f32(ap + base + 0, v.x * norm);
        atom_add_f32(ap + base + 1, v.y * norm);
        atom_add_f32(ap + base + 2, v.z * norm);
        atom_add_f32(ap + base + 3, v.w * norm);
    }
}

// ---------------- bias (+relu) epilogue ------------------------------------
__global__ void k_bias_act(const float* __restrict__ in,
                           const float* __restrict__ bias,
                           float* __restrict__ out, int n, int relu) {
    int i = blockIdx.x * blockDim.x + threadIdx.x;  // over N*F
    if (i < n) {
        float v = in[i] + bias[i & (GF - 1)];
        if (relu) v = fmaxf(v, 0.0f);
        out[i] = v;
    }
}

// ---------------- gather-sum pooling: sent[b,:] = sum_l h[tok[b,l],:] ------
__global__ void k_pool(const float* __restrict__ h,
                       const int* __restrict__ sentence,
                       float* __restrict__ sent) {
    int b = blockIdx.x;   // 0..B-1
    int f = threadIdx.x;  // 0..255
    float acc = 0.0f;
#pragma unroll 5
    for (int l = 0; l < GL; ++l) {
        int idx = sentence[b * GL + l];
        acc += h[(size_t)idx * GF + f];
    }
    sent[(size_t)b * GF + f] = acc;
}

// ---------------- fc3: out[B,2] = h2[B,128] @ W[128,2] + b -----------------
__global__ void k_fc3(const float* __restrict__ h2,
                      const float* __restrict__ w,
                      const float* __restrict__ bias,
                      float* __restrict__ out) {
    int t = blockIdx.x * blockDim.x + threadIdx.x;  // over B*2
    if (t >= GB * 2) return;
    int b = t >> 1;
    int j = t & 1;
    float acc = bias[j];
    const float* row = h2 + (size_t)b * 128;
#pragma unroll 8
    for (int k = 0; k < 128; ++k) acc += row[k] * w[k * 2 + j];
    out[t] = acc;
}

// ---------------------------------------------------------------------------
extern "C" void kernel_launch(void* const* d_in, const int* in_sizes, int n_in,
                              void* d_out, int out_size, void* d_ws, size_t ws_size,
                              hipStream_t stream) {
    (void)in_sizes; (void)n_in; (void)out_size; (void)ws_size;

    const float* x        = (const float*)d_in[0];
    const int*   edge     = (const int*)d_in[1];   // [2, E]
    const int*   sentence = (const int*)d_in[2];   // [B, L]
    const float* W1       = (const float*)d_in[3];
    const float* b1       = (const float*)d_in[4];
    const float* W2       = (const float*)d_in[5];
    const float* b2       = (const float*)d_in[6];
    const float* fc1_w    = (const float*)d_in[7];
    const float* fc1_b    = (const float*)d_in[8];
    const float* fc2_w    = (const float*)d_in[9];
    const float* fc2_b    = (const float*)d_in[10];
    const float* fc3_w    = (const float*)d_in[11];
    const float* fc3_b    = (const float*)d_in[12];
    float* out = (float*)d_out;

    const int* src = edge;
    const int* dst = edge + GE;

    // workspace carve-up (256B aligned)
    char* ws = (char*)d_ws;
    size_t off = 0;
    auto take = [&](size_t bytes) {
        void* p = ws + off;
        off += (bytes + 255) & ~(size_t)255;
        return p;
    };
    float* dinv = (float*)take((size_t)GN * 4);
    float* bufA = (float*)take((size_t)GN * GF * 4);
    float* bufB = (float*)take((size_t)GN * GF * 4);
    float* sent = (float*)take((size_t)GB * GF * 4);
    float* m1   = (float*)take((size_t)GB * 256 * 4);
    float* m2   = (float*)take((size_t)GB * 128 * 4);

    const int NF = GN * GF;

    // 1) symmetric normalization factors
    k_deg_init<<<(GN + 255) / 256, 256, 0, stream>>>(dinv, GN);
    k_deg_accum<<<(GE + 255) / 256, 256, 0, stream>>>(dst, dinv, GE);
    k_deg_to_dinv<<<(GN + 255) / 256, 256, 0, stream>>>(dinv, GN);

    // GEMM grids: Nc=256 -> 2 row-blocks (32 rows) per workgroup;
    //             Nc=128 -> 4 row-blocks (64 rows) per workgroup.
    const int rb256 = 2, rb128 = 4;

    // 2) layer 1: h = x @ W1  (bias folded in after aggregation)
    {
        dim3 grid((GN / 16 + rb256 - 1) / rb256);
        k_wmma_gemm_f32<<<grid, 256, 0, stream>>>(x, W1, nullptr, bufA, GN, GF, GF, 0);
    }
    k_self_init<<<(NF + 255) / 256, 256, 0, stream>>>(bufA, dinv, bufB, NF);
    k_edge_agg<<<(GE + 7) / 8, 256, 0, stream>>>(bufA, src, dst, dinv, bufB, GE);
    k_bias_act<<<(NF + 255) / 256, 256, 0, stream>>>(bufB, b1, bufA, NF, 1);

    // 3) layer 2: t = h1 @ W2
    {
        dim3 grid((GN / 16 + rb256 - 1) / rb256);
        k_wmma_gemm_f32<<<grid, 256, 0, stream>>>(bufA, W2, nullptr, bufB, GN, GF, GF, 0);
    }
    k_self_init<<<(NF + 255) / 256, 256, 0, stream>>>(bufB, dinv, bufA, NF);
    k_edge_agg<<<(GE + 7) / 8, 256, 0, stream>>>(bufB, src, dst, dinv, bufA, GE);
    k_bias_act<<<(NF + 255) / 256, 256, 0, stream>>>(bufA, b2, bufA, NF, 0);

    // 4) gather-sum pooling
    k_pool<<<GB, 256, 0, stream>>>(bufA, sentence, sent);

    // 5) MLP head
    {
        dim3 grid((GB / 16 + rb256 - 1) / rb256);
        k_wmma_gemm_f32<<<grid, 256, 0, stream>>>(sent, fc1_w, fc1_b, m1, GB, 256, 256, 1);
    }
    {
        dim3 grid((GB / 16 + rb128 - 1) / rb128);
        k_wmma_gemm_f32<<<grid, 256, 0, stream>>>(m1, fc2_w, fc2_b, m2, GB, 256, 128, 1);
    }
    k_fc3<<<(GB * 2 + 255) / 256, 256, 0, stream>>>(m2, fc3_w, fc3_b, out);
}